// PathEmbedding_22093311770743
// MI455X (gfx1250) — compile-verified
//
#include <hip/hip_runtime.h>
#include <hip/hip_bf16.h>

#define NUM_QUESTS 8192
#define NUM_PATHS  8
#define LINK_DIM   64
#define PATH_LEN   16
#define NUM_LINKS  10000
#define P_TOTAL    (NUM_QUESTS * NUM_PATHS)   // 65536 paths
#define N_IN       (NUM_LINKS * LINK_DIM)     // 640000 floats

typedef float v2f __attribute__((ext_vector_type(2)));
typedef float v8f __attribute__((ext_vector_type(8)));

// ---------------------------------------------------------------------------
// Pass 1: partial sums of squares (deterministic tree reduction, no atomics)
// ---------------------------------------------------------------------------
__global__ void k_sumsq_partial(const float* __restrict__ x,
                                float* __restrict__ partial, int n) {
    __shared__ float smem[256];
    float s = 0.0f;
    for (int i = blockIdx.x * blockDim.x + threadIdx.x; i < n;
         i += gridDim.x * blockDim.x) {
        float v = x[i];
        s += v * v;
    }
    smem[threadIdx.x] = s;
    __syncthreads();
    for (int off = 128; off > 0; off >>= 1) {
        if ((int)threadIdx.x < off) smem[threadIdx.x] += smem[threadIdx.x + off];
        __syncthreads();
    }
    if (threadIdx.x == 0) partial[blockIdx.x] = smem[0];
}

// Pass 2: fold partials -> scale = rsqrt(max(sumsq, 1e-12))
__global__ void k_finalize_scale(const float* __restrict__ partial, int nblocks,
                                 float* __restrict__ scale_out) {
    __shared__ float smem[256];
    float s = 0.0f;
    for (int i = threadIdx.x; i < nblocks; i += 256) s += partial[i];
    smem[threadIdx.x] = s;
    __syncthreads();
    for (int off = 128; off > 0; off >>= 1) {
        if ((int)threadIdx.x < off) smem[threadIdx.x] += smem[threadIdx.x + off];
        __syncthreads();
    }
    if (threadIdx.x == 0) scale_out[0] = rsqrtf(fmaxf(smem[0], 1e-12f));
}

// ---------------------------------------------------------------------------
// Link projection: y[n][0..2] = (scale*inputs[n]) @ kernel + bias[0]
// One wave (32 lanes) per 16-link tile; 16 chained V_WMMA_F32_16X16X4_F32.
// B fragments (kernel * scale, loop-invariant) are hoisted into registers
// with UNCONDITIONAL clamped-index loads, so the WMMA loop body is just
// one 8B A-load + one WMMA — no EXEC-mask branching, no per-chunk waits.
//   A(16x4)  frag: vgpr j, lane -> A[lane&15][2*(lane>>4)+j]
//   B(4x16)  frag: vgpr j, lane -> B[2*(lane>>4)+j][lane&15]
//   D(16x16) frag: vgpr j, lane -> D[j+8*(lane>>4)][lane&15]
// ---------------------------------------------------------------------------
__global__ void k_project_wmma(const float* __restrict__ inputs,  // 10000x64
                               const float* __restrict__ kern,    // 64x3
                               const float* __restrict__ bias,    // 2x3
                               const float* __restrict__ scale_p,
                               float* __restrict__ y) {           // 10000x4
    const int lane = threadIdx.x;          // one wave per block
    const int tile = blockIdx.x;           // 625 tiles * 16 rows = 10000
    const int m    = lane & 15;
    const int half = lane >> 4;            // 0 or 1
    const int ncol = m;                    // B/D column this lane holds
    const float scale = scale_p[0];

    // Predication folded into the multiplier; address always valid.
    const float bscale = (ncol < 3) ? scale : 0.0f;
    const int   colIdx = (ncol < 3) ? ncol : 0;

    // Hoist all 16 B fragments (2 floats each) into registers.
    float bx[16], by[16];
#pragma unroll
    for (int c = 0; c < 16; ++c) {
        const int k0 = c * 4 + 2 * half;
        bx[c] = bscale * kern[k0 * 3 + colIdx];
        by[c] = bscale * kern[(k0 + 1) * 3 + colIdx];
    }

    const int    row  = tile * 16 + m;
    const float* arow = inputs + row * LINK_DIM;

    v8f acc = {};                          // f32 accumulator, 8 VGPRs
#pragma unroll
    for (int c = 0; c < 16; ++c) {         // K = 64 in chunks of 4
        const int k0 = c * 4 + 2 * half;   // even -> 8B aligned float2 load
        v2f a = *(const v2f*)(arow + k0);
        v2f b;
        b.x = bx[c];
        b.y = by[c];
        acc = __builtin_amdgcn_wmma_f32_16x16x4_f32(
            /*neg_a=*/false, a, /*neg_b=*/false, b,
            /*c_mod=*/(short)0, acc, /*reuse_a=*/false, /*reuse_b=*/false);
    }

    if (ncol < 3) {
        const float b0 = bias[ncol];       // bias[0][k]
#pragma unroll
        for (int j = 0; j < 8; ++j) {
            const int r = tile * 16 + j + 8 * half;
            y[r * 4 + ncol] = acc[j] + b0;
        }
    }
}

// ---------------------------------------------------------------------------
// GRU scan: one thread per path, 16 sequential scalar GRU steps (UNITS==1).
// lens == PATH_LEN for every path, so the (t < lens) mask is identically true.
// All 16 y-table gathers are independent of the recurrence -> issue them ALL
// up-front so the sequential sigmoid/tanh chain runs out of registers and
// pays gather latency once, not 16 times.
// ---------------------------------------------------------------------------
__global__ void k_gru(const float* __restrict__ y,       // 10000x4 table
                      const float* __restrict__ flow,    // P
                      const float* __restrict__ rk,      // 1x3
                      const float* __restrict__ bias,    // 2x3
                      const int*   __restrict__ paths,   // P*16
                      float* __restrict__ out) {         // P  (grid is exact)
    const int p = blockIdx.x * blockDim.x + threadIdx.x;

    const float rk0 = rk[0], rk1 = rk[1], rk2 = rk[2];
    const float b1z = bias[3], b1r = bias[4], b1h = bias[5];

    const int4*   pp = (const int4*)(paths + p * PATH_LEN);
    const float4* yt = (const float4*)y;

    // 16 link ids via four 128b loads.
    int links[PATH_LEN];
#pragma unroll
    for (int q = 0; q < 4; ++q) {
        const int4 li = pp[q];
        links[4 * q + 0] = li.x;
        links[4 * q + 1] = li.y;
        links[4 * q + 2] = li.z;
        links[4 * q + 3] = li.w;
    }

    // Issue all 16 gathers before the recurrence.
    float4 yv[PATH_LEN];
#pragma unroll
    for (int t = 0; t < PATH_LEN; ++t) yv[t] = yt[links[t]];

    float h = flow[p];
#pragma unroll
    for (int t = 0; t < PATH_LEN; ++t) {
        const float z  = 1.0f / (1.0f + __expf(-(yv[t].x + h * rk0 + b1z)));
        const float r  = 1.0f / (1.0f + __expf(-(yv[t].y + h * rk1 + b1r)));
        const float hh = tanhf(yv[t].z + r * (h * rk2 + b1h));
        h = z * h + (1.0f - z) * hh;
    }
    out[p] = h;
}

// ---------------------------------------------------------------------------
extern "C" void kernel_launch(void* const* d_in, const int* in_sizes, int n_in,
                              void* d_out, int out_size, void* d_ws, size_t ws_size,
                              hipStream_t stream) {
    const float* inputs = (const float*)d_in[0];  // (10000, 64)
    const float* flow   = (const float*)d_in[1];  // (65536,)
    const float* kern   = (const float*)d_in[2];  // (64, 3)
    const float* rk     = (const float*)d_in[3];  // (1, 3)
    const float* bias   = (const float*)d_in[4];  // (2, 3)
    const int*   paths  = (const int*)d_in[5];    // (1048576,)
    float* out = (float*)d_out;

    // ws layout (floats): [0] scale | [64..704) partials | [1024..) y table
    float* ws      = (float*)d_ws;
    float* scale   = ws;
    float* partial = ws + 64;
    float* ytab    = ws + 1024;                   // 10000*4 floats = 160 KB

    const int REDUCE_BLOCKS = 640;
    k_sumsq_partial<<<REDUCE_BLOCKS, 256, 0, stream>>>(inputs, partial, N_IN);
    k_finalize_scale<<<1, 256, 0, stream>>>(partial, REDUCE_BLOCKS, scale);
    k_project_wmma<<<NUM_LINKS / 16, 32, 0, stream>>>(inputs, kern, bias, scale, ytab);
    k_gru<<<P_TOTAL / 256, 256, 0, stream>>>(ytab, flow, rk, bias, paths, out);
}